// HarmonicLoss_67216238183251
// MI455X (gfx1250) — compile-verified
//
#include <hip/hip_runtime.h>
#include <math.h>

typedef float v8f __attribute__((ext_vector_type(8)));
typedef float v4f __attribute__((ext_vector_type(4)));
typedef float v2f __attribute__((ext_vector_type(2)));

constexpr int   B_  = 4;
constexpr int   S_  = 512;
constexpr int   D_  = 64;
constexpr int   V_  = 8192;
constexpr float EPS_ = 1e-8f;
constexpr int   OUT_DIM = 20;
constexpr float MAXV = 20.0f;

// ---------- workspace layout (bytes) ----------
// [0      , 16384): u64 packed argmax  [B*S]
// [16384  , 49152): f32 ||u_tgt||^2    [V]
// [49152  , 57344): f32 ||e_tgt||^2    [B*S]
// [57344  , 57376): f32 partial sums   [8]

__device__ __forceinline__ float pow_inv(float sq, float nval) {
    // (sqrt(max(sq,0)) + eps)^(-n)
    float norm = sqrtf(fmaxf(sq, 0.0f));
    return __expf(-nval * __logf(norm + EPS_));
}

__device__ __forceinline__ unsigned long long pack_vi(float val, int v) {
    // val > 0 always (inv * hm of positive terms), so float-bit order == uint order
    return ((unsigned long long)__float_as_uint(val) << 32) | (unsigned)v;
}

__device__ __forceinline__ unsigned long long ullmax(unsigned long long a, unsigned long long b) {
    return a > b ? a : b;
}

__device__ __forceinline__ unsigned long long shfl_xor16_max(unsigned long long x) {
    unsigned lo = (unsigned)x, hi = (unsigned)(x >> 32);
    unsigned lo2 = (unsigned)__shfl_xor((int)lo, 16, 32);
    unsigned hi2 = (unsigned)__shfl_xor((int)hi, 16, 32);
    unsigned long long y = ((unsigned long long)hi2 << 32) | lo2;
    return ullmax(x, y);
}

// ---------- pass 0: zero the atomically-updated scratch ----------
__global__ void init_ws_kernel(unsigned long long* __restrict__ amax,
                               float* __restrict__ partials) {
    int i = blockIdx.x * blockDim.x + threadIdx.x;
    if (i < B_ * S_) amax[i] = 0ull;
    if (i < 8)       partials[i] = 0.0f;
}

// ---------- pass 1: squared norms of target rows ----------
__global__ void norms_kernel(const float* __restrict__ Ut, const float* __restrict__ Et,
                             float* __restrict__ nu, float* __restrict__ ne) {
    int i = blockIdx.x * blockDim.x + threadIdx.x;
    if (i >= V_ + B_ * S_) return;
    const float* row = (i < V_) ? (Ut + (size_t)i * D_) : (Et + (size_t)(i - V_) * D_);
    float acc = 0.0f;
    #pragma unroll
    for (int d = 0; d < D_; d += 4) {
        v4f x = *(const v4f*)(row + d);
        acc += x.x * x.x + x.y * x.y + x.z * x.z + x.w * x.w;
    }
    if (i < V_) nu[i] = acc; else ne[i - V_] = acc;
}

// ---------- pass 2: fused f32 WMMA GEMM + inv/hm epilogue + packed argmax ----------
// grid: (V/16/8) * (S/16) = 64 * 32 = 2048 blocks of 256 threads (8 waves).
// Block -> one s-tile (16 cols), 8 consecutive v-tiles (one per wave).
__global__ __launch_bounds__(256)
void target_argmax_kernel(const float* __restrict__ Et,   // (B,S,D) target
                          const float* __restrict__ Ut,   // (V,D)   target unembedding
                          const float* __restrict__ nu,
                          const float* __restrict__ ne,
                          const float* __restrict__ n_ptr,
                          unsigned long long* __restrict__ amax) {
    const int lane  = threadIdx.x & 31;
    const int wave  = threadIdx.x >> 5;
    const int sTile = blockIdx.x & 31;        // S/16 = 32
    const int vGrp  = blockIdx.x >> 5;        // 64 groups of 8 v-tiles
    const int vbase = (vGrp * 8 + wave) * 16;
    const int sbase = sTile * 16;
    const int m     = lane & 15;              // A: row M ; B: col N (same formula)
    const int khalf = (lane >> 4) << 1;       // 0 for lanes 0-15, 2 for lanes 16-31
    const float nval = n_ptr[0];

    // A row pointer (u) and the 4 B row pointers (e for each b), offset by khalf
    const float* ua = Ut + (size_t)(vbase + m) * D_ + khalf;
    const float* e0 = Et + ((size_t)(0 * S_) + sbase + m) * D_ + khalf;
    const float* e1 = Et + ((size_t)(1 * S_) + sbase + m) * D_ + khalf;
    const float* e2 = Et + ((size_t)(2 * S_) + sbase + m) * D_ + khalf;
    const float* e3 = Et + ((size_t)(3 * S_) + sbase + m) * D_ + khalf;

    v8f c0 = {}, c1 = {}, c2 = {}, c3 = {};
    #pragma unroll
    for (int k = 0; k < 16; ++k) {            // K = 64 in steps of 4
        v2f a  = *(const v2f*)(ua + 4 * k);
        v2f b0 = *(const v2f*)(e0 + 4 * k);
        v2f b1 = *(const v2f*)(e1 + 4 * k);
        v2f b2 = *(const v2f*)(e2 + 4 * k);
        v2f b3 = *(const v2f*)(e3 + 4 * k);
        c0 = __builtin_amdgcn_wmma_f32_16x16x4_f32(false, a, false, b0, (short)0, c0, false, false);
        c1 = __builtin_amdgcn_wmma_f32_16x16x4_f32(false, a, false, b1, (short)0, c1, false, false);
        c2 = __builtin_amdgcn_wmma_f32_16x16x4_f32(false, a, false, b2, (short)0, c2, false, false);
        c3 = __builtin_amdgcn_wmma_f32_16x16x4_f32(false, a, false, b3, (short)0, c3, false, false);
    }

    // Epilogue: element (v,s) of all four accumulators sits in the same lane/slot.
    const int voff = (lane >> 4) << 3;        // C/D layout: M = r (+8 for lanes 16-31)
    const int scol = sbase + m;
    const float neb0 = ne[0 * S_ + scol];
    const float neb1 = ne[1 * S_ + scol];
    const float neb2 = ne[2 * S_ + scol];
    const float neb3 = ne[3 * S_ + scol];

    unsigned long long best0 = 0, best1 = 0, best2 = 0, best3 = 0;
    #pragma unroll
    for (int r = 0; r < 8; ++r) {
        const int v = vbase + voff + r;
        const float nuv = nu[v];
        float i0 = pow_inv(nuv + neb0 - 2.0f * c0[r], nval);
        float i1 = pow_inv(nuv + neb1 - 2.0f * c1[r], nval);
        float i2 = pow_inv(nuv + neb2 - 2.0f * c2[r], nval);
        float i3 = pow_inv(nuv + neb3 - 2.0f * c3[r], nval);
        float hm = 0.25f * (i0 + i1 + i2 + i3);
        best0 = ullmax(best0, pack_vi(i0 * hm, v));
        best1 = ullmax(best1, pack_vi(i1 * hm, v));
        best2 = ullmax(best2, pack_vi(i2 * hm, v));
        best3 = ullmax(best3, pack_vi(i3 * hm, v));
    }

    // Combine the two lane halves holding the same column
    best0 = shfl_xor16_max(best0);
    best1 = shfl_xor16_max(best1);
    best2 = shfl_xor16_max(best2);
    best3 = shfl_xor16_max(best3);

    __shared__ unsigned long long red[8][4][16];
    if (lane < 16) {
        red[wave][0][lane] = best0;
        red[wave][1][lane] = best1;
        red[wave][2][lane] = best2;
        red[wave][3][lane] = best3;
    }
    __syncthreads();

    const int t = threadIdx.x;
    if (t < 64) {
        const int b = t >> 4, col = t & 15;
        unsigned long long mx = red[0][b][col];
        #pragma unroll
        for (int w = 1; w < 8; ++w) mx = ullmax(mx, red[w][b][col]);
        atomicMax(&amax[b * S_ + sbase + col], mx);   // commutative -> deterministic
    }
}

// ---------- pass 3: output-side inv only at argmax index; -log(sel) ----------
__global__ __launch_bounds__(256)
void select_loss_kernel(const float* __restrict__ Eo,   // (B,S,D) output
                        const float* __restrict__ Uo,   // (V,D)   output unembedding
                        const float* __restrict__ n_ptr,
                        const unsigned long long* __restrict__ amax,
                        float* __restrict__ partials) {
    const int i = blockIdx.x * blockDim.x + threadIdx.x;   // 0..2047
    const int b = i >> 9;
    const int s = i & 511;
    const float nval = n_ptr[0];
    const unsigned v = (unsigned)(amax[i] & 0xFFFFFFFFu);
    const float* urow = Uo + (size_t)v * D_;

    float inv[4];
    #pragma unroll
    for (int bp = 0; bp < 4; ++bp) {
        const float* erow = Eo + ((size_t)bp * S_ + s) * D_;
        float acc = 0.0f;
        #pragma unroll
        for (int d = 0; d < D_; d += 4) {
            v4f u4 = *(const v4f*)(urow + d);
            v4f e4 = *(const v4f*)(erow + d);
            float dx = e4.x - u4.x, dy = e4.y - u4.y, dz = e4.z - u4.z, dw = e4.w - u4.w;
            acc += dx * dx + dy * dy + dz * dz + dw * dw;
        }
        inv[bp] = pow_inv(acc, nval);
    }
    const float hm = 0.25f * (inv[0] + inv[1] + inv[2] + inv[3]);
    float part = -__logf(inv[b] * hm);

    __shared__ float sred[256];
    sred[threadIdx.x] = part;
    __syncthreads();
    for (int st = 128; st > 0; st >>= 1) {
        if (threadIdx.x < st) sred[threadIdx.x] += sred[threadIdx.x + st];
        __syncthreads();
    }
    if (threadIdx.x == 0) partials[blockIdx.x] = sred[0];
}

// ---------- pass 4: binomial pmf over 20 bins ----------
__global__ void finalize_kernel(const float* __restrict__ partials, float* __restrict__ out) {
    __shared__ float p_s;
    if (threadIdx.x == 0) {
        float sum = 0.0f;
        for (int j = 0; j < 8; ++j) sum += partials[j];   // serial -> deterministic
        float loss = sum / (float)(B_ * S_);
        float cl = fminf(fmaxf(loss, -MAXV), MAXV);
        p_s = (cl + MAXV) / (2.0f * MAXV);
    }
    __syncthreads();
    const int k = threadIdx.x;
    if (k < OUT_DIM) {
        const float kf = (float)k;
        const float N  = (float)(OUT_DIM - 1);
        const float p  = p_s;
        float logpmf = lgammaf(N + 1.0f) - lgammaf(kf + 1.0f) - lgammaf(N - kf + 1.0f)
                     + kf * logf(p) + (N - kf) * log1pf(-p);
        out[k] = expf(logpmf);
    }
}

extern "C" void kernel_launch(void* const* d_in, const int* in_sizes, int n_in,
                              void* d_out, int out_size, void* d_ws, size_t ws_size,
                              hipStream_t stream) {
    const float* Eo = (const float*)d_in[0];   // output            (B,S,D)
    const float* Uo = (const float*)d_in[1];   // output_unembedding(V,D)
    const float* Et = (const float*)d_in[2];   // target            (B,S,D)
    const float* Ut = (const float*)d_in[3];   // target_unembedding(V,D)
    const float* nP = (const float*)d_in[4];   // n (1,)

    char* ws = (char*)d_ws;
    unsigned long long* amax = (unsigned long long*)(ws);
    float* nu       = (float*)(ws + 16384);
    float* ne       = (float*)(ws + 49152);
    float* partials = (float*)(ws + 57344);
    float* outf     = (float*)d_out;

    init_ws_kernel<<<dim3((B_ * S_) / 256), dim3(256), 0, stream>>>(amax, partials);
    norms_kernel<<<dim3((V_ + B_ * S_ + 255) / 256), dim3(256), 0, stream>>>(Ut, Et, nu, ne);
    target_argmax_kernel<<<dim3((V_ / 16 / 8) * (S_ / 16)), dim3(256), 0, stream>>>(
        Et, Ut, nu, ne, nP, amax);
    select_loss_kernel<<<dim3((B_ * S_) / 256), dim3(256), 0, stream>>>(Eo, Uo, nP, amax, partials);
    finalize_kernel<<<dim3(1), dim3(32), 0, stream>>>(partials, outf);
}